// LUT_iSSAM_20598663152377
// MI455X (gfx1250) — compile-verified
//
#include <hip/hip_runtime.h>
#include <hip/hip_bf16.h>

typedef float v2f __attribute__((ext_vector_type(2)));
typedef float v8f __attribute__((ext_vector_type(8)));

#define HW   (512 * 512)
#define FHW  (32 * 32)
#define LUTC 35937            // 33*33*33
#define LUT3 107811           // 3*35937

// ---- workspace layout (float offsets) ----
#define OFF_DM     0          // 8*1024
#define OFF_MEANDM 8192       // 8 (+pad)
#define OFF_PF     8208       // 16*256 (rows 8..15 zero padding)
#define OFF_PB     12304      // 16*256
#define OFF_FG     16400      // 8*256
#define OFF_BG     18448      // 8*256
#define OFF_W      20496      // 8*4
#define OFF_LUTB   20528      // 8*107811  (~3.45 MB total ws use)

// ---------------------------------------------------------------------------
// 1) Antialiased bilinear 16x downsample of mask (jax.image.resize semantics:
//    triangle kernel stretched by scale=16, edge-renormalized). Also zeroes
//    the padded pf/pb block (8192 floats) used as the WMMA A matrices.
__global__ void k_downmask(const float* __restrict__ mask, float* __restrict__ ws) {
    int tid = blockIdx.x * blockDim.x + threadIdx.x;
    if (tid < 8192) {                         // zero pf..pb (contiguous 8192 floats)
        ws[OFF_PF + tid] = 0.0f;
    }
    if (tid >= 8 * FHW) return;
    int b  = tid >> 10;
    int oy = (tid >> 5) & 31;
    int ox = tid & 31;
    const float* mb = mask + (size_t)b * HW;
    float cy = 16.0f * oy + 7.5f;
    float cx = 16.0f * ox + 7.5f;
    int y0 = 16 * oy - 8, x0 = 16 * ox - 8;

    float sy = 0.0f, sx = 0.0f;
    #pragma unroll
    for (int t = 0; t < 32; ++t) {
        int y = y0 + t;
        float wy = 1.0f - fabsf((float)y - cy) * (1.0f / 16.0f);
        if (y >= 0 && y < 512 && wy > 0.0f) sy += wy;
        int x = x0 + t;
        float wx = 1.0f - fabsf((float)x - cx) * (1.0f / 16.0f);
        if (x >= 0 && x < 512 && wx > 0.0f) sx += wx;
    }
    float acc = 0.0f;
    for (int ty = 0; ty < 32; ++ty) {
        int y = y0 + ty;
        if (y < 0 || y >= 512) continue;
        float wy = 1.0f - fabsf((float)y - cy) * (1.0f / 16.0f);
        if (wy <= 0.0f) continue;
        const float* row = mb + (size_t)y * 512;
        float racc = 0.0f;
        for (int tx = 0; tx < 32; ++tx) {
            int x = x0 + tx;
            if (x < 0 || x >= 512) continue;
            float wx = 1.0f - fabsf((float)x - cx) * (1.0f / 16.0f);
            if (wx > 0.0f) racc += wx * row[x];
        }
        acc += wy * racc;
    }
    ws[OFF_DM + tid] = acc / (sy * sx);
}

// ---------------------------------------------------------------------------
// 2) mean of down_mask per batch. 1 block x 256 threads = 8 waves, wave b.
__global__ void k_meandm(float* __restrict__ ws) {
    int b = threadIdx.x >> 5;
    int lane = threadIdx.x & 31;
    const float* d = ws + OFF_DM + b * FHW;
    float s = 0.0f;
    for (int i = lane; i < FHW; i += 32) s += d[i];
    #pragma unroll
    for (int off = 16; off > 0; off >>= 1) s += __shfl_xor(s, off, 32);
    if (lane == 0) ws[OFF_MEANDM + b] = s * (1.0f / (float)FHW);
}

// ---------------------------------------------------------------------------
// 3) Pooled features: pf[b,c] = mean(enc*dm), pb[b,c] = mean(enc) - pf.
//    One wave per (b,c): 2048 waves = 256 blocks x 8 waves.
__global__ void k_pool(const float* __restrict__ enc, float* __restrict__ ws) {
    int w = blockIdx.x * (blockDim.x >> 5) + (threadIdx.x >> 5);
    int lane = threadIdx.x & 31;
    int b = w >> 8;
    int c = w & 255;
    const float* e = enc + ((size_t)b * 256 + c) * FHW;
    const float* d = ws + OFF_DM + b * FHW;
    float s1 = 0.0f, s0 = 0.0f;
    for (int i = lane; i < FHW; i += 32) {
        float ev = e[i];
        s0 += ev;
        s1 += ev * d[i];
    }
    #pragma unroll
    for (int off = 16; off > 0; off >>= 1) {
        s0 += __shfl_xor(s0, off, 32);
        s1 += __shfl_xor(s1, off, 32);
    }
    if (lane == 0) {
        ws[OFF_PF + b * 256 + c] = s1 * (1.0f / (float)FHW);
        ws[OFF_PB + b * 256 + c] = (s0 - s1) * (1.0f / (float)FHW);
    }
}

// ---------------------------------------------------------------------------
// 4) WMMA GEMM: [16(batch-pad) x 256(k)] @ [256(k) x 256(o)] for fg and bg.
//    V_WMMA_F32_16X16X4_F32, 16 waves (one 16-wide o-tile each), 64 k-steps,
//    two accumulators sharing the B matrix -> 128 v_wmma per wave.
__global__ void k_gemm_wmma(const float* __restrict__ conv_w,
                            const float* __restrict__ conv_b,
                            float* __restrict__ ws) {
    int wave = threadIdx.x >> 5;        // 0..15 -> o tile
    int lane = threadIdx.x & 31;
    int nb = wave * 16;
    int r  = lane & 15;                 // A row (M) / B col (N) within tile
    int kh = (lane >> 4) << 1;          // +0 for lanes 0-15, +2 for 16-31

    const float* pA  = ws + OFF_PF + r * 256 + kh;
    const float* pAb = ws + OFF_PB + r * 256 + kh;
    const float* pB  = conv_w + (size_t)(nb + r) * 256 + kh;

    v8f accf = {0.f, 0.f, 0.f, 0.f, 0.f, 0.f, 0.f, 0.f};
    v8f accb = {0.f, 0.f, 0.f, 0.f, 0.f, 0.f, 0.f, 0.f};

    for (int k0 = 0; k0 < 256; k0 += 4) {
        v2f a  = *(const v2f*)(pA + k0);
        v2f ab = *(const v2f*)(pAb + k0);
        v2f bm = *(const v2f*)(pB + k0);
        accf = __builtin_amdgcn_wmma_f32_16x16x4_f32(
            false, a,  false, bm, (short)0, accf, false, false);
        accb = __builtin_amdgcn_wmma_f32_16x16x4_f32(
            false, ab, false, bm, (short)0, accb, false, false);
    }

    // D layout: VGPR i, lanes 0-15 -> (M=i, N=lane). Lanes 16-31 hold padded
    // batches 8..15 -> discard.
    if (lane < 16) {
        int o = nb + lane;
        float cb = conv_b[o];
        #pragma unroll
        for (int i = 0; i < 8; ++i) {
            float md = ws[OFF_MEANDM + i];
            ws[OFF_FG + i * 256 + o] = accf[i] + cb * md;
            ws[OFF_BG + i * 256 + o] = accb[i] + cb * (1.0f - md);
        }
    }
}

// ---------------------------------------------------------------------------
// 5) weights[b,n] = [fg|bg] . fc_w[n,:] + fc_b[n].  32 dots of length 512.
__global__ void k_weights(const float* __restrict__ fc_w,
                          const float* __restrict__ fc_b,
                          float* __restrict__ ws) {
    int lane = threadIdx.x;
    if (lane >= 32) return;
    int b = lane >> 2, n = lane & 3;
    const float* wf = fc_w + n * 512;
    const float* fg = ws + OFF_FG + b * 256;
    const float* bg = ws + OFF_BG + b * 256;
    float s = fc_b[n];
    for (int o = 0; o < 256; ++o) s += wf[o] * fg[o] + wf[256 + o] * bg[o];
    ws[OFF_W + lane] = s;
}

// ---------------------------------------------------------------------------
// 6) Per-batch combined LUT: lutb[b] = sum_n w[b,n] * lut_n.  (L2 resident)
__global__ void k_lutmix(const float* __restrict__ l0, const float* __restrict__ l1,
                         const float* __restrict__ l2, const float* __restrict__ l3,
                         float* __restrict__ ws) {
    int i = blockIdx.x * blockDim.x + threadIdx.x;
    int b = blockIdx.y;
    if (i >= LUT3) return;
    const float* w = ws + OFF_W + b * 4;
    float w0 = w[0], w1 = w[1], w2 = w[2], w3 = w[3];
    ws[OFF_LUTB + (size_t)b * LUT3 + i] =
        w0 * l0[i] + w1 * l1[i] + w2 * l2[i] + w3 * l3[i];
}

// ---------------------------------------------------------------------------
// 7) Trilinear LUT apply + mask composite. One thread per pixel.
__global__ void k_apply(const float* __restrict__ image,
                        const float* __restrict__ mask,
                        const float* __restrict__ ws,
                        float* __restrict__ out) {
    int p = blockIdx.x * blockDim.x + threadIdx.x;
    int b = blockIdx.y;
    if (p >= HW) return;
    const float* img = image + (size_t)b * 3 * HW;
    float r  = img[p];
    float g  = img[HW + p];
    float bl = img[2 * HW + p];
    float m  = mask[(size_t)b * HW + p];

    float sr = r * 32.0f, sg = g * 32.0f, sb = bl * 32.0f;
    int ir = min(max((int)floorf(sr), 0), 31);
    int ig = min(max((int)floorf(sg), 0), 31);
    int ib = min(max((int)floorf(sb), 0), 31);
    float fr = sr - (float)ir, fgc = sg - (float)ig, fb = sb - (float)ib;
    float wr0 = 1.0f - fr, wg0 = 1.0f - fgc, wb0 = 1.0f - fb;

    float w000 = wb0 * wg0 * wr0, w001 = wb0 * wg0 * fr;
    float w010 = wb0 * fgc * wr0, w011 = wb0 * fgc * fr;
    float w100 = fb  * wg0 * wr0, w101 = fb  * wg0 * fr;
    float w110 = fb  * fgc * wr0, w111 = fb  * fgc * fr;

    const float* L = ws + OFF_LUTB + (size_t)b * LUT3 + ib * 1089 + ig * 33 + ir;
    float inv = 1.0f - m;
    #pragma unroll
    for (int c = 0; c < 3; ++c) {
        const float* Lc = L + c * LUTC;
        float v = Lc[0]    * w000 + Lc[1]    * w001
                + Lc[33]   * w010 + Lc[34]   * w011
                + Lc[1089] * w100 + Lc[1090] * w101
                + Lc[1122] * w110 + Lc[1123] * w111;
        float in_c = (c == 0) ? r : ((c == 1) ? g : bl);
        out[(size_t)b * 3 * HW + (size_t)c * HW + p] = v * m + in_c * inv;
    }
}

// ---------------------------------------------------------------------------
extern "C" void kernel_launch(void* const* d_in, const int* in_sizes, int n_in,
                              void* d_out, int out_size, void* d_ws, size_t ws_size,
                              hipStream_t stream) {
    const float* enc_feat = (const float*)d_in[0];
    const float* image    = (const float*)d_in[1];
    const float* mask     = (const float*)d_in[2];
    const float* conv_w   = (const float*)d_in[3];
    const float* conv_b   = (const float*)d_in[4];
    const float* fc_w     = (const float*)d_in[5];
    const float* fc_b     = (const float*)d_in[6];
    const float* lut0     = (const float*)d_in[7];
    const float* lut1     = (const float*)d_in[8];
    const float* lut2     = (const float*)d_in[9];
    const float* lut3     = (const float*)d_in[10];
    float* out = (float*)d_out;
    float* ws  = (float*)d_ws;
    (void)in_sizes; (void)n_in; (void)out_size; (void)ws_size;

    // 1) mask downsample + zero padded WMMA A-matrix rows
    k_downmask<<<32, 256, 0, stream>>>(mask, ws);
    // 2) per-batch mean of down_mask
    k_meandm<<<1, 256, 0, stream>>>(ws);
    // 3) pooled features (wave per (b,c))
    k_pool<<<256, 256, 0, stream>>>(enc_feat, ws);
    // 4) WMMA f32 GEMM -> fg/bg features
    k_gemm_wmma<<<1, 512, 0, stream>>>(conv_w, conv_b, ws);
    // 5) 8x4 LUT blending weights
    k_weights<<<1, 32, 0, stream>>>(fc_w, fc_b, ws);
    // 6) combined per-batch LUTs
    {
        dim3 grid((LUT3 + 255) / 256, 8);
        k_lutmix<<<grid, 256, 0, stream>>>(lut0, lut1, lut2, lut3, ws);
    }
    // 7) trilerp + composite
    {
        dim3 grid(HW / 256, 8);
        k_apply<<<grid, 256, 0, stream>>>(image, mask, ws, out);
    }
}